// SphericalHarmonics_34127810134278
// MI455X (gfx1250) — compile-verified
//
#include <hip/hip_runtime.h>
#include <cstdint>
#include <cstddef>

// ---------------------------------------------------------------------------
// Real spherical harmonics up to L_MAX=8, sphericart ordering, output split
// into per-l blocks:  out[N*l^2 + i*(2l+1) + j],  j = l+m.
// Streaming, store-bandwidth-bound: 12B in / 324B out per point (~336 MB/call
// -> ~14.4 us roofline at 23.3 TB/s; ~4e8 FLOPs total, far below any compute
// ceiling). Strategy: compute per-thread via ALP recurrence -> stage in LDS
// already in output order -> coalesced b128 writeback via CDNA5 async
// store-from-LDS (ASYNCcnt path, LDS->L2 direct, no VGPR round-trip).
// ---------------------------------------------------------------------------

#define TPB  128
#define LMAXK 8
#define NCOL  81   // (LMAX+1)^2

// ---------------- compile-time constant tables -----------------------------
constexpr double cfact(int n) { double r = 1.0; for (int i = 2; i <= n; ++i) r *= (double)i; return r; }
constexpr double csqrt(double x) { double g = x > 1.0 ? x : 1.0; for (int i = 0; i < 100; ++i) g = 0.5 * (g + x / g); return g; }

struct KTab {
    float k[9][9];    // k[l][m]: norm (with sqrt(2) folded in for m>0)
    float pmm[9];     // (2m-1)!! seeds
};

constexpr KTab make_ktab() {
    KTab t{};
    const double PI = 3.14159265358979323846;
    const double S2 = csqrt(2.0);
    for (int m = 0; m <= LMAXK; ++m) {
        double pm = 1.0;
        for (int i = 1; i < 2 * m; i += 2) pm *= (double)i;
        t.pmm[m] = (float)pm;
        for (int l = m; l <= LMAXK; ++l) {
            double nrm = csqrt(((2.0 * l + 1.0) / (4.0 * PI)) * (cfact(l - m) / cfact(l + m)));
            if (m > 0) nrm *= S2;
            t.k[l][m] = (float)nrm;
        }
    }
    return t;
}
constexpr KTab KT = make_ktab();

typedef float v4f __attribute__((ext_vector_type(4)));
// The async builtins take pointers to 16-byte int vectors in explicit
// address spaces (per hipcc's diagnostic of the builtin's signature).
typedef int v4i __attribute__((__vector_size__(4 * sizeof(int))));
typedef v4i __attribute__((address_space(1)))* gptr_v4i;
typedef v4i __attribute__((address_space(3)))* lptr_v4i;

// ---------------- CDNA5 async store-from-LDS detection ---------------------
#ifndef SH_ASYNC
#if defined(__has_builtin)
#if __has_builtin(__builtin_amdgcn_global_store_async_from_lds_b128)
#define SH_ASYNC 1
#endif
#endif
#endif
#ifndef SH_ASYNC
#define SH_ASYNC 0
#endif

__global__ __launch_bounds__(TPB) void
sph_l8_kernel(const float* __restrict__ R, float* __restrict__ out, int n)
{
    // LDS staged exactly in output order for this block's 128 points:
    // region for degree l starts at TPB*l*l floats; point t's row at t*(2l+1).
    __shared__ __align__(16) float sh[TPB * NCOL];

    const int tid = threadIdx.x;
    const int i0  = blockIdx.x * TPB;
    const int i   = i0 + tid;

    if (i < n) {
        // Warm L2 for the next block's input (guarded, non-speculative safe).
        if (i + TPB < n) __builtin_prefetch(R + 3 * (size_t)(i + TPB), 0, 1);

        float x = R[3 * (size_t)i + 0];
        float y = R[3 * (size_t)i + 1];
        float z = R[3 * (size_t)i + 2];
        float rinv = rsqrtf(x * x + y * y + z * z);
        x *= rinv; y *= rinv; z *= rinv;

        // Per-l LDS row bases, centered at j=l (so cols are rb[l]+-m).
        int rb[9];
#pragma unroll
        for (int l = 0; l <= LMAXK; ++l)
            rb[l] = TPB * l * l + tid * (2 * l + 1) + l;

        float A = 1.0f, B = 0.0f;   // cos(m*phi), sin(m*phi) (unnormalized)
#pragma unroll
        for (int m = 0; m <= LMAXK; ++m) {
            if (m > 0) { float t = x * A - y * B; B = x * B + y * A; A = t; }
            float p1 = 0.0f, p2 = 0.0f;
#pragma unroll
            for (int l = m; l <= LMAXK; ++l) {
                float p;
                if (l == m)          p = KT.pmm[m];
                else if (l == m + 1) p = (2.0f * (float)m + 1.0f) * z * p1;
                else                 p = ((2.0f * (float)l - 1.0f) * z * p1
                                          - (float)(l + m - 1) * p2) * (1.0f / (float)(l - m));
                const float np = KT.k[l][m] * p;
                if (m == 0) {
                    sh[rb[l]] = np;
                } else {
                    sh[rb[l] + m] = np * A;
                    sh[rb[l] - m] = np * B;
                }
                p2 = p1; p1 = p;
            }
        }
    }
    __syncthreads();

    // -------- coalesced writeback: per-l regions are contiguous ------------
    int cnt = n - i0; if (cnt > TPB) cnt = TPB;
    const size_t nsz = (size_t)n;

    if (cnt == TPB && ((n & 3) == 0)) {
        // full block, 16B-aligned: b128 per lane
#pragma unroll
        for (int l = 0; l <= LMAXK; ++l) {
            const int w = 2 * l + 1;
            float* dst = out + (size_t)(l * l) * nsz + (size_t)i0 * (size_t)w;
            const float* src = sh + TPB * l * l;
            const int nvec = (TPB / 4) * w;              // float4 chunks in region
            for (int k = tid; k < nvec; k += TPB) {
#if SH_ASYNC
                // CDNA5 async LDS->global store (ASYNCcnt path, no VGPR round-trip)
                __builtin_amdgcn_global_store_async_from_lds_b128(
                    (gptr_v4i)(dst + 4 * k),
                    (lptr_v4i)(src + 4 * k),
                    0, 0);
#else
                v4f v = *(const v4f*)(src + 4 * k);
                __builtin_nontemporal_store(v, (v4f*)(dst + 4 * k));
#endif
            }
        }
    } else {
        // ragged tail block: scalar coalesced copies
#pragma unroll
        for (int l = 0; l <= LMAXK; ++l) {
            const int w = 2 * l + 1;
            float* dst = out + (size_t)(l * l) * nsz + (size_t)i0 * (size_t)w;
            const float* src = sh + TPB * l * l;
            const int lim = cnt * w;
            for (int k = tid; k < lim; k += TPB)
                __builtin_nontemporal_store(src[k], dst + k);
        }
    }

#if SH_ASYNC
    // Fence outstanding async LDS->global traffic before the wave retires
    // (S_ENDPGM also implies wait-idle, but be explicit).
#if defined(__has_builtin) && __has_builtin(__builtin_amdgcn_s_wait_asynccnt)
    __builtin_amdgcn_s_wait_asynccnt(0);
#else
    asm volatile("s_wait_asynccnt 0" ::: "memory");
#endif
#endif
}

extern "C" void kernel_launch(void* const* d_in, const int* in_sizes, int n_in,
                              void* d_out, int out_size, void* d_ws, size_t ws_size,
                              hipStream_t stream)
{
    (void)n_in; (void)out_size; (void)d_ws; (void)ws_size;
    const float* R = (const float*)d_in[0];
    float* out = (float*)d_out;
    const int n = in_sizes[0] / 3;                 // R is [N,3]
    const int grid = (n + TPB - 1) / TPB;
    sph_l8_kernel<<<grid, TPB, 0, stream>>>(R, out, n);
}